// NeuralCDE_44470091382934
// MI455X (gfx1250) — compile-verified
//
#include <hip/hip_runtime.h>
#include <hip/hip_bf16.h>
#include <math.h>

// Shapes from the reference
#define LL   141
#define CCH  8      // C (input channels)
#define HH   64     // hidden
#define MIDD 128    // mid layer
#define HCC  512    // H*C

typedef __attribute__((ext_vector_type(2))) float v2f;
typedef __attribute__((ext_vector_type(8))) float v8f;

// f32 WMMA 16x16x4:  D(16x16,f32) = A(16x4,f32) * B(4x16,f32) + C
#define WMMA4(a, b, c) \
  __builtin_amdgcn_wmma_f32_16x16x4_f32(false, (a), false, (b), (short)0, (c), false, false)

// A fragment: A is 16(M) x 4(K) chunk at `base` (row stride ld), K-base kb.
// Layout (ISA 7.12.2, 32-bit A 16x4): lanes 0-15 -> K={kb,kb+1}, lanes 16-31 -> K={kb+2,kb+3}
__device__ __forceinline__ v2f load_a4(const float* base, int ld, int kb, int lane) {
  int m = lane & 15;
  int k = kb + ((lane >> 4) << 1);
  const float* p = base + m * ld + k;
  v2f r = {p[0], p[1]};
  return r;
}

// B fragment where logical B[k][n] = W[n][k] (W row-major (N,K), row stride ldw).
// Mirrors A layout: lane indexes n, half-wave splits K.
__device__ __forceinline__ v2f load_bT(const float* W, int ldw, int n0, int kb, int lane) {
  int n = n0 + (lane & 15);
  int k = kb + ((lane >> 4) << 1);
  const float* p = W + n * ldw + k;
  v2f r = {p[0], p[1]};
  return r;
}

// g(t,z):  h=relu(z@W1^T+b1); f=tanh(h@W2^T+b2) viewed (16,H,C); k=einsum(f,dX)
// z: LDS 16xHH, h: LDS 16xMIDD scratch, kout: LDS 16xHH, dX: LDS 16xCCH
__device__ __forceinline__ void g_eval(const float* z, float* h, float* kout,
                                       const float* dX,
                                       const float* __restrict__ W1, const float* __restrict__ b1,
                                       const float* __restrict__ W2, const float* __restrict__ b2,
                                       int lane) {
  // ---- h = relu(z @ W1^T + b1) : (16,64)@(64,128)
  // Hoist all A fragments of z (loop-invariant across the 8 N-tiles): 16 x v2f = 32 VGPRs
  v2f aZ[HH / 4];
#pragma unroll
  for (int ks = 0; ks < HH / 4; ++ks) aZ[ks] = load_a4(z, HH, ks * 4, lane);

  for (int nt = 0; nt < MIDD / 16; ++nt) {
    v8f acc = {};
#pragma unroll
    for (int ks = 0; ks < HH / 4; ++ks) {
      v2f b = load_bT(W1, HH, nt * 16, ks * 4, lane);
      acc = WMMA4(aZ[ks], b, acc);
    }
    int col = nt * 16 + (lane & 15);
    float bias = b1[col];
#pragma unroll
    for (int j = 0; j < 8; ++j) {
      int row = (lane < 16) ? j : j + 8;
      h[row * MIDD + col] = fmaxf(acc[j] + bias, 0.0f);
    }
  }
  __syncthreads();

  // ---- f tiles + contraction with dX/dt
  int cc   = lane & 7;          // c index within h-block (col & 7)
  int hsel = (lane >> 3) & 1;   // which of the 2 h columns in this 16-wide tile

  // Hoist all A fragments of h (invariant across 32 N-tiles): 32 x v2f = 64 VGPRs
  v2f aH[MIDD / 4];
#pragma unroll
  for (int ks = 0; ks < MIDD / 4; ++ks) aH[ks] = load_a4(h, MIDD, ks * 4, lane);

  // dX values are also invariant across N-tiles: one per j
  float dxv[8];
#pragma unroll
  for (int j = 0; j < 8; ++j) {
    int row = (lane < 16) ? j : j + 8;
    dxv[j] = dX[row * CCH + cc];
  }

  for (int nt = 0; nt < HCC / 16; ++nt) {
    v8f acc = {};
#pragma unroll
    for (int ks = 0; ks < MIDD / 4; ++ks) {
      v2f b = load_bT(W2, MIDD, nt * 16, ks * 4, lane);
      acc = WMMA4(aH[ks], b, acc);
    }
    int col  = nt * 16 + (lane & 15);
    float bias = b2[col];
    int hout = nt * 2 + hsel;
#pragma unroll
    for (int j = 0; j < 8; ++j) {
      int row = (lane < 16) ? j : j + 8;
      float f = tanhf(acc[j] + bias);
      float p = f * dxv[j];
      p += __shfl_xor(p, 1, 32);   // reduce over the 8 c-lanes
      p += __shfl_xor(p, 2, 32);
      p += __shfl_xor(p, 4, 32);
      if (cc == 0) kout[row * HH + hout] = p;
    }
  }
  __syncthreads();
}

__global__ __launch_bounds__(32) void ncde_fused(
    const float* __restrict__ coeffs,
    const float* __restrict__ W_init, const float* __restrict__ b_init,
    const float* __restrict__ W1, const float* __restrict__ b1,
    const float* __restrict__ W2, const float* __restrict__ b2,
    const float* __restrict__ W_out, const float* __restrict__ b_out,
    float* __restrict__ out) {
  __shared__ float s_z [16 * HH];
  __shared__ float s_z0[16 * HH];
  __shared__ float s_h [16 * MIDD];
  __shared__ float s_k1[16 * HH];
  __shared__ float s_k2[16 * HH];
  __shared__ float s_kc[16 * HH];
  __shared__ float s_ac[16 * HH];
  __shared__ float s_dX[3][16 * CCH];

  const int lane = threadIdx.x;
  const int r0   = blockIdx.x * 16;   // batch-row base for this wave
  const float dt = 140.0f / 49.0f;

  // dX/dt at knot indices 0,1,2 (the only ones one RK4 step touches)
  for (int i = 0; i < 3; ++i)
    for (int q = lane; q < 16 * CCH; q += 32) {
      int b = q >> 3, c = q & 7;
      long base = ((long)(r0 + b) * LL + i) * CCH + c;
      s_dX[i][q] = coeffs[base + CCH] - coeffs[base];
    }

  // z0 = X(0) @ W_init^T + b_init : (16,8)@(8,64)
  {
    v2f aX[CCH / 4];
#pragma unroll
    for (int ks = 0; ks < CCH / 4; ++ks)
      aX[ks] = load_a4(coeffs + (long)r0 * LL * CCH, LL * CCH, ks * 4, lane);
    for (int nt = 0; nt < HH / 16; ++nt) {
      v8f acc = {};
#pragma unroll
      for (int ks = 0; ks < CCH / 4; ++ks) {
        v2f b = load_bT(W_init, CCH, nt * 16, ks * 4, lane);
        acc = WMMA4(aX[ks], b, acc);
      }
      int col = nt * 16 + (lane & 15);
      float bias = b_init[col];
#pragma unroll
      for (int j = 0; j < 8; ++j) {
        int row = (lane < 16) ? j : j + 8;
        float v = acc[j] + bias;
        s_z0[row * HH + col] = v;
        s_z [row * HH + col] = v;
      }
    }
  }
  __syncthreads();

  // ---- RK4 (3/8 rule), single step t: 0 -> dt (only zs[0] is consumed upstream)
  // k1 @ i=0
  g_eval(s_z, s_h, s_kc, s_dX[0], W1, b1, W2, b2, lane);
  for (int q = lane; q < 16 * HH; q += 32) {
    float k1 = s_kc[q];
    s_k1[q] = k1;
    s_ac[q] = k1;
    s_z[q]  = s_z0[q] + dt * (k1 * (1.0f / 3.0f));
  }
  __syncthreads();

  // k2 @ t=dt/3 -> i=0
  g_eval(s_z, s_h, s_kc, s_dX[0], W1, b1, W2, b2, lane);
  for (int q = lane; q < 16 * HH; q += 32) {
    float k2 = s_kc[q];
    s_k2[q] = k2;
    s_ac[q] += 3.0f * k2;
    s_z[q]  = s_z0[q] + dt * (k2 - s_k1[q] * (1.0f / 3.0f));
  }
  __syncthreads();

  // k3 @ t=2dt/3 -> i=1
  g_eval(s_z, s_h, s_kc, s_dX[1], W1, b1, W2, b2, lane);
  for (int q = lane; q < 16 * HH; q += 32) {
    float k3 = s_kc[q];
    s_ac[q] += 3.0f * k3;
    s_z[q]  = s_z0[q] + dt * (s_k1[q] - s_k2[q] + k3);
  }
  __syncthreads();

  // k4 @ t=dt -> i=2 ; z1 = z0 + (k1 + 3k2 + 3k3 + k4)*dt/8
  g_eval(s_z, s_h, s_kc, s_dX[2], W1, b1, W2, b2, lane);
  for (int q = lane; q < 16 * HH; q += 32) {
    s_z[q] = s_z0[q] + (s_ac[q] + s_kc[q]) * (dt * 0.125f);
  }
  __syncthreads();

  // out = z1 @ W_out^T + b_out : (16,64)@(64,2); one (b,o) pair per lane
  {
    int b = lane & 15, o = lane >> 4;
    float s = b_out[o];
    for (int hh = 0; hh < HH; ++hh)
      s = fmaf(s_z[b * HH + hh], W_out[o * HH + hh], s);
    out[(long)(r0 + b) * 2 + o] = s;
  }
}

extern "C" void kernel_launch(void* const* d_in, const int* in_sizes, int n_in,
                              void* d_out, int out_size, void* d_ws, size_t ws_size,
                              hipStream_t stream) {
  (void)n_in; (void)out_size; (void)d_ws; (void)ws_size;
  const float* coeffs = (const float*)d_in[0];
  const float* W_init = (const float*)d_in[1];
  const float* b_init = (const float*)d_in[2];
  const float* W1     = (const float*)d_in[3];
  const float* b1     = (const float*)d_in[4];
  const float* W2     = (const float*)d_in[5];
  const float* b2     = (const float*)d_in[6];
  const float* W_out  = (const float*)d_in[7];
  const float* b_out  = (const float*)d_in[8];
  float* out = (float*)d_out;

  int B = in_sizes[0] / (LL * CCH);   // 4096
  int blocks = B / 16;                // one wave (32 threads) per 16 batch rows
  ncde_fused<<<blocks, 32, 0, stream>>>(coeffs, W_init, b_init, W1, b1, W2, b2,
                                        W_out, b_out, out);
}